// GraphLayer_39530878993007
// MI455X (gfx1250) — compile-verified
//
#include <hip/hip_runtime.h>
#include <hip/hip_bf16.h>

typedef __attribute__((ext_vector_type(2))) float v2f;
typedef __attribute__((ext_vector_type(8))) float v8f;

#define LN_EPS 1e-3f
#define WSTRIDE 80     // weight LDS row stride (floats): 16-bank offset between k and k+1 rows
#define ISTRIDE 196    // 192-wide activation tile, stride%64==4 -> conflict-free A-frag reads
#define XSTRIDE 324    // 320-wide activation tile, stride%64==4

// ---------------------------------------------------------------------------
// Kernel A: attention branch -> V_dst_out = V_dst + relu(o @ Wo + bo)
// grid = 32 (b*8+d), block = 128 threads (4 waves; wave w owns N cols [16w,16w+16))
// ---------------------------------------------------------------------------
__global__ __launch_bounds__(128)
void attn_kernel(const float* __restrict__ Vsrc, const float* __restrict__ Vdst,
                 const float* __restrict__ E,    const float* __restrict__ A,
                 const float* __restrict__ g_vd, const float* __restrict__ b_vd,
                 const float* __restrict__ g_in, const float* __restrict__ b_in,
                 const float* __restrict__ Wq,   const float* __restrict__ bq,
                 const float* __restrict__ Wk,   const float* __restrict__ bk,
                 const float* __restrict__ Wv,   const float* __restrict__ bv,
                 const float* __restrict__ Wo,   const float* __restrict__ bo,
                 float* __restrict__ outV)
{
    extern __shared__ float sm[];
    float* wkl  = sm;                      // 192*80
    float* wvl  = wkl  + 192 * WSTRIDE;    // 192*80
    float* inp  = wvl  + 192 * WSTRIDE;    // 16*196 (LayerNormed activations)
    float* klds = inp  + 16 * ISTRIDE;     // 16*64
    float* vlds = klds + 16 * 64;          // 16*64
    float* qb   = vlds + 16 * 64;          // 64
    float* ob   = qb   + 64;               // 64 accumulated o
    float* vn   = ob   + 64;               // 128
    float* mu   = vn   + 128;              // 16
    float* rsd  = mu   + 16;               // 16
    float* sbuf = rsd  + 16;               // 128 scores
    float* wbuf = sbuf + 128;              // 128 softmax weights

    const int tid  = threadIdx.x;
    const int lane = tid & 31;
    const int wave = tid >> 5;
    const int pair = blockIdx.x;           // b*8 + d
    const int b    = pair >> 3;
    const int d    = pair & 7;

    // stage Wk, Wv into LDS (padded stride)
    for (int i = tid; i < 192 * 64; i += 128) {
        int r = i >> 6, c = i & 63;
        wkl[r * WSTRIDE + c] = Wk[i];
        wvl[r * WSTRIDE + c] = Wv[i];
    }

    // --- LN(V_dst row) -> vn, then q = relu(vn @ Wq + bq) ---
    float xv = 0.f;
    if (tid < 128) { xv = Vdst[pair * 128 + tid]; vn[tid] = xv; }
    __syncthreads();
    if (tid == 0) {
        float s = 0.f, s2 = 0.f;
        for (int i = 0; i < 128; ++i) { float x = vn[i]; s += x; s2 += x * x; }
        float m = s * (1.f / 128.f);
        mu[0]  = m;
        rsd[0] = rsqrtf(s2 * (1.f / 128.f) - m * m + LN_EPS);
    }
    __syncthreads();
    float m0 = mu[0], r0v = rsd[0];
    if (tid < 128) vn[tid] = (xv - m0) * r0v * g_vd[tid] + b_vd[tid];
    __syncthreads();
    if (tid < 64) {
        float acc = bq[tid];
        for (int i = 0; i < 128; ++i) acc += vn[i] * Wq[i * 64 + tid];
        qb[tid] = fmaxf(acc, 0.f);
        ob[tid] = 0.f;
    }
    __syncthreads();

    const float* Arow = A + b * 2048 + d;  // A[b,s,d] = Arow[s*8]

    const int mrow = lane & 15;
    const int ksel = (lane >> 4) * 2;
    const int ncol = wave * 16 + (lane & 15);
    const int mb   = (lane >> 4) * 8;
    const float bkb = bk[ncol], bvb = bv[ncol];

    for (int st = 0; st < 16; ++st) {
        const int s0 = st * 16;
        // build raw inp tile: concat(Vsrc*A, E), rows = s0..s0+15
        for (int i = tid; i < 16 * 192; i += 128) {
            int row = i / 192, col = i - row * 192;
            int s = s0 + row;
            float av = Arow[s * 8];
            float val = (col < 128)
                      ? Vsrc[(b * 256 + s) * 128 + col] * av
                      : E[((b * 256 + s) * 8 + d) * 64 + (col - 128)];
            inp[row * ISTRIDE + col] = val;
        }
        __syncthreads();
        // per-row LayerNorm stats
        if (tid < 16) {
            const float* rp = inp + tid * ISTRIDE;
            float s = 0.f, s2 = 0.f;
            for (int i = 0; i < 192; ++i) { float x = rp[i]; s += x; s2 += x * x; }
            float m = s * (1.f / 192.f);
            mu[tid]  = m;
            rsd[tid] = rsqrtf(s2 * (1.f / 192.f) - m * m + LN_EPS);
        }
        __syncthreads();
        for (int i = tid; i < 16 * 192; i += 128) {
            int row = i / 192, col = i - row * 192;
            float x = inp[row * ISTRIDE + col];
            inp[row * ISTRIDE + col] = (x - mu[row]) * rsd[row] * g_in[col] + b_in[col];
        }
        __syncthreads();

        // WMMA: k = relu(inpn @ Wk + bk), val = relu(inpn @ Wv + bv) (16x192x64, f32)
        v8f ck = {}; v8f cv = {};
        for (int kk = 0; kk < 192; kk += 4) {
            v2f a, bkf, bvf;
            a.x   = inp[mrow * ISTRIDE + kk + ksel];
            a.y   = inp[mrow * ISTRIDE + kk + ksel + 1];
            bkf.x = wkl[(kk + ksel) * WSTRIDE + ncol];
            bkf.y = wkl[(kk + ksel + 1) * WSTRIDE + ncol];
            bvf.x = wvl[(kk + ksel) * WSTRIDE + ncol];
            bvf.y = wvl[(kk + ksel + 1) * WSTRIDE + ncol];
            ck = __builtin_amdgcn_wmma_f32_16x16x4_f32(false, a, false, bkf, (short)0, ck, false, false);
            cv = __builtin_amdgcn_wmma_f32_16x16x4_f32(false, a, false, bvf, (short)0, cv, false, false);
        }
        for (int r = 0; r < 8; ++r) {
            klds[(mb + r) * 64 + ncol] = fmaxf(ck[r] + bkb, 0.f);
            vlds[(mb + r) * 64 + ncol] = fmaxf(cv[r] + bvb, 0.f);
        }
        __syncthreads();

        // score[row,h] = (q[h,:].k[row,h,:]) / sqrt(8)
        {
            int row = tid >> 3, h = tid & 7;
            float acc = 0.f;
            for (int j = 0; j < 8; ++j) acc += qb[h * 8 + j] * klds[row * 64 + h * 8 + j];
            sbuf[row * 8 + h] = acc * 0.35355339059327373f;
        }
        __syncthreads();
        // softmax over heads per row
        if (tid < 16) {
            float mx = sbuf[tid * 8];
            for (int h = 1; h < 8; ++h) mx = fmaxf(mx, sbuf[tid * 8 + h]);
            float e[8], sum = 0.f;
            for (int h = 0; h < 8; ++h) { e[h] = __expf(sbuf[tid * 8 + h] - mx); sum += e[h]; }
            float inv = 1.f / sum;
            for (int h = 0; h < 8; ++h) wbuf[tid * 8 + h] = e[h] * inv;
        }
        __syncthreads();
        // o[h*8+j] += sum_rows w[row,h] * val[row,h*8+j]
        if (tid < 64) {
            int h = tid >> 3;
            float acc = 0.f;
            for (int r = 0; r < 16; ++r) acc += wbuf[r * 8 + h] * vlds[r * 64 + tid];
            ob[tid] += acc;
        }
        __syncthreads();
    }

    // V_dst_out = V_dst + relu(o @ Wo + bo)
    if (tid < 128) {
        float acc = bo[tid];
        for (int i = 0; i < 64; ++i) acc += ob[i] * Wo[i * 128 + tid];
        outV[pair * 128 + tid] = Vdst[pair * 128 + tid] + fmaxf(acc, 0.f);
    }
}

// ---------------------------------------------------------------------------
// Kernel B: E_new = relu(concat(Vsrc*A, Vdst*A, E) @ We + be), A_new = 1
// grid = 128 WGs, each strides over 512 row-tiles of 16; We loaded to LDS once
// ---------------------------------------------------------------------------
__global__ __launch_bounds__(128)
void edge_kernel(const float* __restrict__ Vsrc, const float* __restrict__ Vdst,
                 const float* __restrict__ E,    const float* __restrict__ A,
                 const float* __restrict__ We,   const float* __restrict__ be,
                 float* __restrict__ Eout, float* __restrict__ Aout)
{
    extern __shared__ float sm[];
    float* wel = sm;                       // 320*80
    float* xt  = sm + 320 * WSTRIDE;       // 16*324

    const int tid  = threadIdx.x;
    const int lane = tid & 31;
    const int wave = tid >> 5;

    for (int i = tid; i < 320 * 64; i += 128) {
        int r = i >> 6, c = i & 63;
        wel[r * WSTRIDE + c] = We[i];
    }
    __syncthreads();

    const int mrow = lane & 15;
    const int ksel = (lane >> 4) * 2;
    const int ncol = wave * 16 + (lane & 15);
    const int mb   = (lane >> 4) * 8;
    const float beb = be[ncol];

    for (int tile = blockIdx.x; tile < 512; tile += gridDim.x) {
        const int r0 = tile * 16;
        for (int i = tid; i < 16 * 320; i += 128) {
            int row = i / 320, col = i - row * 320;
            int rr = r0 + row;             // flat (b,s,d)
            int bb = rr >> 11;
            int ss = (rr >> 3) & 255;
            int dd = rr & 7;
            float av = A[rr];
            float val;
            if      (col < 128) val = Vsrc[(bb * 256 + ss) * 128 + col] * av;
            else if (col < 256) val = Vdst[(bb * 8 + dd) * 128 + (col - 128)] * av;
            else                val = E[rr * 64 + (col - 256)];
            xt[row * XSTRIDE + col] = val;
        }
        if (tid < 16) Aout[r0 + tid] = 1.0f;   // softmax over singleton -> ones
        __syncthreads();

        v8f acc8 = {};
        for (int kk = 0; kk < 320; kk += 4) {
            v2f a, bw;
            a.x  = xt[mrow * XSTRIDE + kk + ksel];
            a.y  = xt[mrow * XSTRIDE + kk + ksel + 1];
            bw.x = wel[(kk + ksel) * WSTRIDE + ncol];
            bw.y = wel[(kk + ksel + 1) * WSTRIDE + ncol];
            acc8 = __builtin_amdgcn_wmma_f32_16x16x4_f32(false, a, false, bw, (short)0, acc8, false, false);
        }
        for (int r = 0; r < 8; ++r)
            Eout[(r0 + mb + r) * 64 + ncol] = fmaxf(acc8[r] + beb, 0.f);
        __syncthreads();
    }
}

extern "C" void kernel_launch(void* const* d_in, const int* in_sizes, int n_in,
                              void* d_out, int out_size, void* d_ws, size_t ws_size,
                              hipStream_t stream) {
    const float* Vsrc = (const float*)d_in[0];
    const float* Vdst = (const float*)d_in[1];
    const float* E    = (const float*)d_in[2];
    const float* A    = (const float*)d_in[3];
    const float* g_vd = (const float*)d_in[4];
    const float* b_vd = (const float*)d_in[5];
    const float* g_in = (const float*)d_in[6];
    const float* b_in = (const float*)d_in[7];
    const float* Wq   = (const float*)d_in[8];
    const float* bq   = (const float*)d_in[9];
    const float* Wk   = (const float*)d_in[10];
    const float* bk   = (const float*)d_in[11];
    const float* Wv   = (const float*)d_in[12];
    const float* bv   = (const float*)d_in[13];
    const float* Wo   = (const float*)d_in[14];
    const float* bo   = (const float*)d_in[15];
    const float* We   = (const float*)d_in[16];
    const float* be   = (const float*)d_in[17];
    // d_in[18]=Wa, d_in[19]=ba unused: softmax over singleton axis is identically 1.

    float* outV = (float*)d_out;            // [4,8,128]      = 4096
    float* Eout = outV + 4096;              // [4,256,8,64]   = 524288
    float* Aout = Eout + 524288;            // [4,256,8]      = 8192

    size_t smA = (size_t)(2 * 192 * WSTRIDE + 16 * ISTRIDE + 2 * 16 * 64
                          + 64 + 64 + 128 + 16 + 16 + 128 + 128) * sizeof(float);
    size_t smB = (size_t)(320 * WSTRIDE + 16 * XSTRIDE) * sizeof(float);

    attn_kernel<<<32, 128, smA, stream>>>(Vsrc, Vdst, E, A, g_vd, b_vd, g_in, b_in,
                                          Wq, bq, Wk, bk, Wv, bv, Wo, bo, outV);
    edge_kernel<<<128, 128, smB, stream>>>(Vsrc, Vdst, E, A, We, be, Eout, Aout);
}